// CLUTNet_9955734192478
// MI455X (gfx1250) — compile-verified
//
#include <hip/hip_runtime.h>
#include <hip/hip_bf16.h>
#include <cstdint>

// ---------------- constants ----------------
#define LDIM 33
#define LD2  1089      // 33*33
#define LD3  35937     // 33^3
#define NLUT 107811    // 3*33^3  (columns of the big GEMM)
#define NUML 20
#define SL   5

typedef __attribute__((ext_vector_type(2))) float v2f;
typedef __attribute__((ext_vector_type(8))) float v8f;

// ---------------- conv0: fused (img-mean)/std, 3->16, s2 p1, + lrelu --------
__global__ void k_conv0(const float* __restrict__ in, const float* __restrict__ w,
                        const float* __restrict__ bias, float* __restrict__ out) {
    // in: 8x3x256x256 ; out: 8x16x128x128
    const int Hin = 256, Win = 256, Hout = 128, Wout = 128, Cout = 16;
    const float sc[3] = {1.f/0.229f, 1.f/0.224f, 1.f/0.225f};
    const float sh[3] = {-0.485f/0.229f, -0.456f/0.224f, -0.406f/0.225f};
    long idx = (long)blockIdx.x * blockDim.x + threadIdx.x;
    long total = 8L * Cout * Hout * Wout;
    if (idx >= total) return;
    int wo = idx % Wout; long t = idx / Wout;
    int ho = t % Hout; t /= Hout;
    int co = t % Cout; int n = t / Cout;
    float acc = bias[co];
    int hi0 = ho * 2 - 1, wi0 = wo * 2 - 1;
    const float* wb = w + (long)co * 3 * 9;
    const float* ib = in + (long)n * 3 * Hin * Win;
#pragma unroll
    for (int ci = 0; ci < 3; ++ci) {
        const float* ip = ib + (long)ci * Hin * Win;
        const float* wp = wb + ci * 9;
#pragma unroll
        for (int kh = 0; kh < 3; ++kh) {
            int hi = hi0 + kh;
            if ((unsigned)hi >= (unsigned)Hin) continue;
#pragma unroll
            for (int kw = 0; kw < 3; ++kw) {
                int wi = wi0 + kw;
                if ((unsigned)wi >= (unsigned)Win) continue;
                float v = ip[hi * Win + wi] * sc[ci] + sh[ci];
                acc += v * wp[kh * 3 + kw];
            }
        }
    }
    out[idx] = acc >= 0.f ? acc : 0.2f * acc;   // fused lrelu
}

// ---------------- generic stride-2 3x3 conv + lrelu ----------------
template <int CIN>
__global__ void k_conv(const float* __restrict__ in, const float* __restrict__ w,
                       const float* __restrict__ bias, float* __restrict__ out,
                       int Cout, int Hin, int Win) {
    int Hout = Hin >> 1, Wout = Win >> 1;
    long idx = (long)blockIdx.x * blockDim.x + threadIdx.x;
    long total = 8L * Cout * Hout * Wout;
    if (idx >= total) return;
    int wo = idx % Wout; long t = idx / Wout;
    int ho = t % Hout; t /= Hout;
    int co = t % Cout; int n = t / Cout;
    float acc = bias[co];
    int hi0 = ho * 2 - 1, wi0 = wo * 2 - 1;
    const float* wb = w + (long)co * CIN * 9;
    const float* ib = in + (long)n * CIN * Hin * Win;
    for (int ci = 0; ci < CIN; ++ci) {
        const float* ip = ib + (long)ci * Hin * Win;
        const float* wp = wb + ci * 9;
#pragma unroll
        for (int kh = 0; kh < 3; ++kh) {
            int hi = hi0 + kh;
            if ((unsigned)hi >= (unsigned)Hin) continue;
#pragma unroll
            for (int kw = 0; kw < 3; ++kw) {
                int wi = wi0 + kw;
                if ((unsigned)wi >= (unsigned)Win) continue;
                acc += ip[hi * Win + wi] * wp[kh * 3 + kw];
            }
        }
    }
    out[idx] = acc >= 0.f ? acc : 0.2f * acc;
}

// ---------------- instance norm (in-place), block per (n,c) plane ----------
__global__ void k_inorm(float* __restrict__ x, const float* __restrict__ g,
                        const float* __restrict__ b, int C, int HW) {
    int plane = blockIdx.x;            // n*C + c
    int c = plane % C;
    float* p = x + (long)plane * HW;
    __shared__ float s1[256], s2[256];
    float sum = 0.f, sq = 0.f;
    for (int i = threadIdx.x; i < HW; i += blockDim.x) {
        float v = p[i]; sum += v; sq += v * v;
    }
    s1[threadIdx.x] = sum; s2[threadIdx.x] = sq;
    __syncthreads();
    for (int s = 128; s > 0; s >>= 1) {
        if ((int)threadIdx.x < s) { s1[threadIdx.x] += s1[threadIdx.x + s]; s2[threadIdx.x] += s2[threadIdx.x + s]; }
        __syncthreads();
    }
    float m = s1[0] / (float)HW;
    float var = s2[0] / (float)HW - m * m;
    float inv = rsqrtf(var + 1e-5f);
    float gg = g[c], bb = b[c];
    for (int i = threadIdx.x; i < HW; i += blockDim.x)
        p[i] = (p[i] - m) * inv * gg + bb;
}

// ---------------- global average pool over 8x8, block per (n,c) ----------
__global__ void k_gap(const float* __restrict__ x, float* __restrict__ feat) {
    int plane = blockIdx.x;            // n*128 + c
    __shared__ float s[64];
    float v = x[(long)plane * 64 + threadIdx.x];
    s[threadIdx.x] = v;
    __syncthreads();
    for (int st = 32; st > 0; st >>= 1) {
        if ((int)threadIdx.x < st) s[threadIdx.x] += s[threadIdx.x + st];
        __syncthreads();
    }
    if (threadIdx.x == 0) feat[plane] = s[0] * (1.f / 64.f);
}

// ---------------- FC: hardswish(feat@W0^T+b0) @ W1^T + b1, padded 16x20 ----
__global__ void k_fc(const float* __restrict__ feat,
                     const float* __restrict__ w0, const float* __restrict__ b0,
                     const float* __restrict__ w1, const float* __restrict__ b1,
                     float* __restrict__ wpad /*16x20, rows>=8 zero*/) {
    __shared__ float h[8 * 128];
    for (int t = threadIdx.x; t < 8 * 128; t += blockDim.x) {
        int n = t / 128, j = t % 128;
        float acc = b0[j];
        const float* f = feat + n * 128;
        const float* wr = w0 + j * 128;
        for (int k = 0; k < 128; ++k) acc += f[k] * wr[k];
        float r = acc + 3.f;
        r = r < 0.f ? 0.f : (r > 6.f ? 6.f : r);
        h[t] = acc * r * (1.f / 6.f);
    }
    __syncthreads();
    for (int t = threadIdx.x; t < 16 * NUML; t += blockDim.x) {
        int n = t / NUML, j = t % NUML;
        float acc = 0.f;
        if (n < 8) {
            acc = b1[j];
            const float* hr = h + n * 128;
            const float* wr = w1 + j * 128;
            for (int k = 0; k < 128; ++k) acc += hr[k] * wr[k];
        }
        wpad[t] = acc;
    }
}

// ---------------- tmp = luts(300x20) @ w_layers(20x1089) ----------
__global__ void k_tmp(const float* __restrict__ luts, const float* __restrict__ wl,
                      float* __restrict__ tmp) {
    int idx = blockIdx.x * blockDim.x + threadIdx.x;
    if (idx >= 300 * LD2) return;
    int c = idx % LD2, r = idx / LD2;
    float acc = 0.f;
#pragma unroll
    for (int k = 0; k < 20; ++k) acc += luts[r * 20 + k] * wl[k * LD2 + c];
    tmp[idx] = acc;
}

// ---------------- fused cube build + _cube_to_lut permutation ----------
// d3luts[n][c][x][y][z] with (i,j,k): c=0 -> (y,z,x); c=1 -> (y,x,z); c=2 -> (x,y,z)
// cube[n,c,i,j,k] = sum_s s_layers[i,s] * tmp[(s*60 + n*3+c)*1089 + j*33+k]
__global__ void k_d3luts(const float* __restrict__ slay, const float* __restrict__ tmp,
                         float* __restrict__ d3luts) {
    long o = (long)blockIdx.x * blockDim.x + threadIdx.x;
    if (o >= (long)NUML * NLUT) return;
    int z = o % LDIM; long t = o / LDIM;
    int y = t % LDIM; t /= LDIM;
    int x = t % LDIM; t /= LDIM;
    int c = t % 3; int n = t / 3;
    int i, j, k;
    if (c == 0)      { i = y; j = z; k = x; }
    else if (c == 1) { i = y; j = x; k = z; }
    else             { i = x; j = y; k = z; }
    int nc = n * 3 + c, jk = j * LDIM + k;
    float acc = 0.f;
#pragma unroll
    for (int s = 0; s < SL; ++s)
        acc += slay[i * SL + s] * tmp[(long)(s * 60 + nc) * LD2 + jk];
    d3luts[o] = acc;
}

// ---------------- d3lut = weight(8x20) @ d3luts(20x107811) via fp32 WMMA ----
// One wave per 16-column tile.  A = padded weights 16x20, K split into 5 x4.
__global__ void k_mixlut_wmma(const float* __restrict__ wpad /*16x20*/,
                              const float* __restrict__ d3luts /*20 x NLUT*/,
                              float* __restrict__ d3lut /*8 x NLUT*/) {
    const int N = NLUT;
    int tile = blockIdx.x * (blockDim.x >> 5) + (threadIdx.x >> 5);
    if (tile * 16 >= N) return;                // wave-uniform: EXEC stays all-1
    int lane = threadIdx.x & 31;
    int half = lane >> 4;                      // 0: K pair (0,1) ; 1: K pair (2,3)
    int l16  = lane & 15;
    int col  = tile * 16 + l16;
    bool colok = col < N;
    int colc = colok ? col : 0;
    v8f acc = {};
#pragma unroll
    for (int ks = 0; ks < 5; ++ks) {
        int k0 = ks * 4 + half * 2;
        v2f a, b;
        a[0] = wpad[l16 * 20 + k0];
        a[1] = wpad[l16 * 20 + k0 + 1];
        float b0v = d3luts[(long)k0 * N + colc];
        float b1v = d3luts[(long)(k0 + 1) * N + colc];
        b[0] = colok ? b0v : 0.f;
        b[1] = colok ? b1v : 0.f;
        acc = __builtin_amdgcn_wmma_f32_16x16x4_f32(
            /*neg_a=*/false, a, /*neg_b=*/false, b,
            /*c_mod=*/(short)0, acc, /*reuse_a=*/false, /*reuse_b=*/false);
    }
    // lanes 0..15 hold rows M=0..7 in acc[0..7]; rows 8..15 (lanes 16..31) are padding
    if (half == 0 && colok) {
#pragma unroll
        for (int r = 0; r < 8; ++r)
            d3lut[(long)r * N + col] = acc[r];
    }
}

// ---------------- trilinear LUT apply + residual, float4 vectorized ----------
__device__ __forceinline__ void trilerp_px(const float* __restrict__ lut,
                                           float r, float g, float b,
                                           float& o0, float& o1, float& o2) {
    const float binsize = (float)(1.000001 / 32.0);
    float pr = r / binsize, pg = g / binsize, pb = b / binsize;
    int r0 = (int)floorf(pr); r0 = r0 < 0 ? 0 : (r0 > 31 ? 31 : r0);
    int g0 = (int)floorf(pg); g0 = g0 < 0 ? 0 : (g0 > 31 ? 31 : g0);
    int b0 = (int)floorf(pb); b0 = b0 < 0 ? 0 : (b0 > 31 ? 31 : b0);
    float rd = pr - (float)r0, gd = pg - (float)g0, bd = pb - (float)b0;
    float rm = 1.f - rd, gm = 1.f - gd, bm = 1.f - bd;
    float w000 = rm * gm * bm, w100 = rd * gm * bm, w010 = rm * gd * bm, w001 = rm * gm * bd;
    float w110 = rd * gd * bm, w101 = rd * gm * bd, w011 = rm * gd * bd, w111 = rd * gd * bd;
    int base = b0 * LD2 + g0 * LDIM + r0;
    float res[3];
#pragma unroll
    for (int c = 0; c < 3; ++c) {
        const float* lc = lut + c * LD3;
        res[c] = lc[base] * w000 + lc[base + 1] * w100 + lc[base + LDIM] * w010 +
                 lc[base + LD2] * w001 + lc[base + LDIM + 1] * w110 +
                 lc[base + LD2 + 1] * w101 + lc[base + LD2 + LDIM] * w011 +
                 lc[base + LD2 + LDIM + 1] * w111;
    }
    o0 = res[0]; o1 = res[1]; o2 = res[2];
}

__global__ void k_trilerp(const float* __restrict__ img_org,
                          const float* __restrict__ d3lut,
                          float* __restrict__ out) {
    const int HW = 720 * 1280;
    const int Q = HW / 4;                      // float4 quads per channel
    long idx = (long)blockIdx.x * blockDim.x + threadIdx.x;
    if (idx >= 8L * Q) return;
    int n = idx / Q, q = idx % Q;
    const float4* ip = (const float4*)(img_org + (long)n * 3 * HW);
    float4 R = ip[q], G = ip[Q + q], B = ip[2 * Q + q];
    const float* lut = d3lut + (long)n * 3 * LD3;
    float4 OR_, OG_, OB_;
    float a, bb, c;
    trilerp_px(lut, R.x, G.x, B.x, a, bb, c); OR_.x = a + R.x; OG_.x = bb + G.x; OB_.x = c + B.x;
    trilerp_px(lut, R.y, G.y, B.y, a, bb, c); OR_.y = a + R.y; OG_.y = bb + G.y; OB_.y = c + B.y;
    trilerp_px(lut, R.z, G.z, B.z, a, bb, c); OR_.z = a + R.z; OG_.z = bb + G.z; OB_.z = c + B.z;
    trilerp_px(lut, R.w, G.w, B.w, a, bb, c); OR_.w = a + R.w; OG_.w = bb + G.w; OB_.w = c + B.w;
    float4* op = (float4*)(out + (long)n * 3 * HW);
    op[q] = OR_; op[Q + q] = OG_; op[2 * Q + q] = OB_;
}

// ---------------- launcher ----------------
extern "C" void kernel_launch(void* const* d_in, const int* in_sizes, int n_in,
                              void* d_out, int out_size, void* d_ws, size_t ws_size,
                              hipStream_t stream) {
    const float* img     = (const float*)d_in[0];
    const float* img_org = (const float*)d_in[1];
    const float* c0w = (const float*)d_in[2];  const float* c0b = (const float*)d_in[3];
    const float* c1w = (const float*)d_in[4];  const float* c1b = (const float*)d_in[5];
    const float* c2w = (const float*)d_in[6];  const float* c2b = (const float*)d_in[7];
    const float* c3w = (const float*)d_in[8];  const float* c3b = (const float*)d_in[9];
    const float* c4w = (const float*)d_in[10]; const float* c4b = (const float*)d_in[11];
    const float* n0g = (const float*)d_in[12]; const float* n0b = (const float*)d_in[13];
    const float* n1g = (const float*)d_in[14]; const float* n1b = (const float*)d_in[15];
    const float* n2g = (const float*)d_in[16]; const float* n2b = (const float*)d_in[17];
    const float* n3g = (const float*)d_in[18]; const float* n3b = (const float*)d_in[19];
    const float* cls0w = (const float*)d_in[20]; const float* cls0b = (const float*)d_in[21];
    const float* cls1w = (const float*)d_in[22]; const float* cls1b = (const float*)d_in[23];
    const float* slay = (const float*)d_in[24];
    const float* wlay = (const float*)d_in[25];
    const float* luts = (const float*)d_in[26];

    float* ws = (float*)d_ws;
    float* x0     = ws;                      // 8*16*128*128 = 2097152
    float* x1     = x0 + 2097152;            // 8*32*64*64   = 1048576
    float* x2     = x1 + 1048576;            // 8*64*32*32   = 524288
    float* x3     = x2 + 524288;             // 8*128*16*16  = 262144
    float* x4     = x3 + 262144;             // 8*128*8*8    = 65536
    float* feat   = x4 + 65536;              // 1024
    float* wpad   = feat + 1024;             // 16*20 = 320
    float* tmp    = wpad + 320;              // 300*1089 = 326700
    float* d3luts = tmp + 326700;            // 20*107811 = 2156220
    float* d3lut  = d3luts + 2156220;        // 8*107811 = 862488

    auto cdiv = [](long a, long b) { return (int)((a + b - 1) / b); };

    // encoder
    k_conv0<<<cdiv(8L*16*128*128, 256), 256, 0, stream>>>(img, c0w, c0b, x0);
    k_inorm<<<8*16, 256, 0, stream>>>(x0, n0g, n0b, 16, 128*128);
    k_conv<16><<<cdiv(8L*32*64*64, 256), 256, 0, stream>>>(x0, c1w, c1b, x1, 32, 128, 128);
    k_inorm<<<8*32, 256, 0, stream>>>(x1, n1g, n1b, 32, 64*64);
    k_conv<32><<<cdiv(8L*64*32*32, 256), 256, 0, stream>>>(x1, c2w, c2b, x2, 64, 64, 64);
    k_inorm<<<8*64, 256, 0, stream>>>(x2, n2g, n2b, 64, 32*32);
    k_conv<64><<<cdiv(8L*128*16*16, 256), 256, 0, stream>>>(x2, c3w, c3b, x3, 128, 32, 32);
    k_inorm<<<8*128, 256, 0, stream>>>(x3, n3g, n3b, 128, 16*16);
    k_conv<128><<<cdiv(8L*128*8*8, 256), 256, 0, stream>>>(x3, c4w, c4b, x4, 128, 16, 16);

    // head
    k_gap<<<8*128, 64, 0, stream>>>(x4, feat);
    k_fc<<<1, 256, 0, stream>>>(feat, cls0w, cls0b, cls1w, cls1b, wpad);

    // LUT construction
    k_tmp<<<cdiv(300L*LD2, 256), 256, 0, stream>>>(luts, wlay, tmp);
    k_d3luts<<<cdiv((long)NUML*NLUT, 256), 256, 0, stream>>>(slay, tmp, d3luts);

    // per-image LUT mix via fp32 WMMA (one wave / 16-col tile, 8 waves / block)
    int ntiles = cdiv(NLUT, 16);
    k_mixlut_wmma<<<cdiv(ntiles, 8), 256, 0, stream>>>(wpad, d3luts, d3lut);

    // trilinear apply + residual
    k_trilerp<<<cdiv(8L * (720*1280/4), 256), 256, 0, stream>>>(img_org, d3lut, (float*)d_out);
}